// SPSN_17540646436867
// MI455X (gfx1250) — compile-verified
//
#include <hip/hip_runtime.h>
#include <cstdint>
#include <cstddef>

typedef __attribute__((ext_vector_type(2))) float v2f;
typedef __attribute__((ext_vector_type(8))) float v8f;

namespace {
constexpr int kB = 16;
constexpr int kT = 2048;
constexpr int kH = 512;
constexpr int kSeg = 32;               // segments along T
constexpr int kSegLen = kT / kSeg;     // 64 timesteps per segment
constexpr int kChunks = kSegLen / 16;  // 4 WMMA chunks per segment
constexpr float kThresh = 1.0f;
constexpr float kSlope = 10.0f;
constexpr float kMaxP = 0.8f;
}  // namespace

__device__ __forceinline__ float clip_beta(const float* bp) {
  return fminf(fmaxf(bp[0], 0.0f), 1.0f);
}

// Cheap counter-based uniform in [0,1): 32-bit low-bias hash of (seed, idx).
// ~8 VALU ops vs ~25 for splitmix64 (which legalizes to v_mul_u64 chains).
__device__ __forceinline__ float rng01(uint32_t seed, uint32_t idx) {
  uint32_t z = idx ^ (seed * 0x9E3779B9u);
  z ^= z >> 16;
  z *= 0x7FEB352Du;
  z ^= z >> 15;
  z *= 0x846CA68Bu;
  z ^= z >> 16;
  return (float)(z >> 8) * (1.0f / 16777216.0f);
}

// Phase 1: per-(b, seg, h) segment tail with zero carry-in.
// c_seg = sum_{s in seg} (1-b) b^(last - s) x[s]
__global__ void spsn_seg_tail(const float* __restrict__ x,
                              const float* __restrict__ beta,
                              float* __restrict__ seg_last) {
  const float b = clip_beta(beta);
  const float omb = 1.0f - b;
  const int gid = blockIdx.x * blockDim.x + threadIdx.x;  // (bb, seg, h)
  const int h = gid % kH;
  const int seg = (gid / kH) % kSeg;
  const int bb = gid / (kH * kSeg);
  const float* xp = x + ((size_t)(bb * kT + seg * kSegLen) * kH + h);
  float c = 0.0f;
#pragma unroll 8
  for (int t = 0; t < kSegLen; ++t) {
    c = fmaf(b, c, omb * xp[(size_t)t * kH]);
  }
  seg_last[gid] = c;
}

// Phase 2: per-(b, h) prefix scan over segment tails with gain b^64.
__global__ void spsn_carry_scan(const float* __restrict__ seg_last,
                                const float* __restrict__ beta,
                                float* __restrict__ carry_in) {
  const float b = clip_beta(beta);
  float bL = b;  // b^64 by 6 squarings
#pragma unroll
  for (int i = 0; i < 6; ++i) bL = bL * bL;
  const int gid = blockIdx.x * blockDim.x + threadIdx.x;  // (bb, h)
  const int h = gid % kH;
  const int bb = gid / kH;
  float c = 0.0f;
#pragma unroll
  for (int s = 0; s < kSeg; ++s) {
    const size_t i = (size_t)(bb * kSeg + s) * kH + h;
    carry_in[i] = c;
    c = fmaf(bL, c, seg_last[i]);
  }
}

// Phase 3: one wave per (b, seg, 16 h-columns). Intra-chunk causal filter as
// a 16x16x16 matmul via 4 chained V_WMMA_F32_16X16X4_F32, carry folded in with
// b^(t+1) factors, spikes sampled per element.
__global__ void spsn_filter_spike(const float* __restrict__ x,
                                  const float* __restrict__ beta,
                                  const int* __restrict__ seedp,
                                  const float* __restrict__ carry_in,
                                  float* __restrict__ spikes,
                                  float* __restrict__ vout) {
  const float b = clip_beta(beta);
  const float omb = 1.0f - b;
  const uint32_t seed = (uint32_t)seedp[0];

  const int lane = threadIdx.x & 31;
  const int lid = lane & 15;   // A-row / B-col / D-col
  const int half = lane >> 4;  // lane half selects K / row group
  const int wave = (blockIdx.x * blockDim.x + threadIdx.x) >> 5;
  const int ht = wave % (kH / 16);
  const int seg = (wave / (kH / 16)) % kSeg;
  const int bb = wave / ((kH / 16) * kSeg);
  const int h0 = ht * 16;

  // bp[d] = b^d
  float bp[16];
  bp[0] = 1.0f;
#pragma unroll
  for (int i = 1; i < 16; ++i) bp[i] = bp[i - 1] * b;

  // A fragments: M[r][s] = (r>=s) ? (1-b) b^(r-s) : 0
  // A 16x4 layout: lane%16 = row r; VGPR0 holds K = k0 + 2*half, VGPR1 = +1.
  v2f a[4];
#pragma unroll
  for (int k = 0; k < 4; ++k) {
    const int s0 = 4 * k + 2 * half;
    a[k].x = (lid >= s0) ? omb * bp[lid - s0] : 0.0f;
    a[k].y = (lid >= s0 + 1) ? omb * bp[lid - s0 - 1] : 0.0f;
  }

  // Carry factors b^(t+1) for row t = v + 8*half (D layout: VGPR v -> row v + 8*half)
  float fac[8];
#pragma unroll
  for (int v = 0; v < 8; ++v) fac[v] = bp[v + 8 * half] * b;

  float carry = carry_in[(size_t)(bb * kSeg + seg) * kH + h0 + lid];

  const uint32_t base = (uint32_t)(bb * kT) * kH + (uint32_t)(h0 + lid);

  for (int ch = 0; ch < kChunks; ++ch) {
    const int t0 = seg * kSegLen + ch * 16;
    v8f acc = {};
#pragma unroll
    for (int k = 0; k < 4; ++k) {
      // B 4x16 layout: lane%16 = col; VGPR0 holds row k0 + 2*half, VGPR1 = +1.
      const int r0 = 4 * k + 2 * half;
      v2f bx;
      bx.x = x[base + (uint32_t)(t0 + r0) * kH];
      bx.y = x[base + (uint32_t)(t0 + r0 + 1) * kH];
      acc = __builtin_amdgcn_wmma_f32_16x16x4_f32(
          /*neg_a=*/false, a[k], /*neg_b=*/false, bx,
          /*c_mod=*/(short)0, acc, /*reuse_a=*/false, /*reuse_b=*/false);
    }
    float v15 = 0.0f;
#pragma unroll
    for (int v = 0; v < 8; ++v) {
      const int t = t0 + v + 8 * half;
      const float Vv = acc[v] + fac[v] * carry;
      const uint32_t idx = base + (uint32_t)t * kH;
      vout[idx] = Vv;
      // spike iff u < MAX_P / (1 + e)  <=>  u*(1+e) < MAX_P   (division-free)
      const float e = __expf(-kSlope * (Vv - kThresh));
      const float u = rng01(seed, idx);
      spikes[idx] = (fmaf(u, e, u) < kMaxP) ? 1.0f : 0.0f;
      if (v == 7) v15 = Vv;
    }
    // New carry = V[15][col], held by lanes 16..31 in VGPR 7.
    carry = __shfl(v15, 16 + lid, 32);
  }
}

extern "C" void kernel_launch(void* const* d_in, const int* in_sizes, int n_in,
                              void* d_out, int out_size, void* d_ws, size_t ws_size,
                              hipStream_t stream) {
  (void)in_sizes; (void)n_in; (void)out_size; (void)ws_size;
  const float* x = (const float*)d_in[0];
  const float* beta = (const float*)d_in[1];
  const int* seed = (const int*)d_in[2];

  float* spikes = (float*)d_out;                // output 0
  float* vout = spikes + (size_t)kB * kT * kH;  // output 1

  float* seg_last = (float*)d_ws;                       // kB*kSeg*kH floats (1 MiB)
  float* carry_in = seg_last + (size_t)kB * kSeg * kH;  // kB*kSeg*kH floats (1 MiB)

  // Phase 1: 16*32*512 = 262144 threads
  spsn_seg_tail<<<(kB * kSeg * kH) / 256, 256, 0, stream>>>(x, beta, seg_last);
  // Phase 2: 16*512 = 8192 threads
  spsn_carry_scan<<<(kB * kH) / 256, 256, 0, stream>>>(seg_last, beta, carry_in);
  // Phase 3: 16*32*32 = 16384 waves, 8 waves per block
  const int waves = kB * kSeg * (kH / 16);
  spsn_filter_spike<<<(waves * 32) / 256, 256, 0, stream>>>(x, beta, seed, carry_in,
                                                            spikes, vout);
}